// CornerPosFeatExtraxtor_49443663511569
// MI455X (gfx1250) — compile-verified
//
#include <hip/hip_runtime.h>
#include <hip/hip_bf16.h>

typedef __attribute__((ext_vector_type(2))) float v2f;
typedef __attribute__((ext_vector_type(8))) float v8f;

#define NB    8
#define CIN   256
#define DD    256
#define HH    128
#define WW    128
#define HWSZ  (HH * WW)
#define MAXR  64
#define MAXC  64
#define GMAX  (2 * MAXR * 2 * MAXC)   // 16384 padded grid points per batch

// ---------------------------------------------------------------------------
// Kernel 0: zero padded output, build mask
// ---------------------------------------------------------------------------
__global__ __launch_bounds__(256) void init_out_kernel(
    const long long* __restrict__ num_rows, const long long* __restrict__ num_cols,
    float* __restrict__ out, float* __restrict__ mask) {
  size_t idx = (size_t)blockIdx.x * blockDim.x + threadIdx.x;
  const size_t total = (size_t)NB * DD * MAXR * MAXC;
  if (idx < total) out[idx] = 0.0f;
  if (idx < (size_t)NB * MAXR * MAXC) {
    int bi = (int)(idx / (MAXR * MAXC));
    int rc = (int)(idx & (MAXR * MAXC - 1));
    int r = rc >> 6, c = rc & 63;
    mask[idx] = (r < (int)num_rows[bi] && c < (int)num_cols[bi]) ? 1.0f : 0.0f;
  }
}

// ---------------------------------------------------------------------------
// Kernel 1: grid-sample + pos-LN + projection GEMM (fp32 WMMA), writes fp_ws
//   fp_ws layout: [bi][p][e], p = py*(2C)+px over the 2R x 2C grid, e in [0,256)
// ---------------------------------------------------------------------------
__global__ __launch_bounds__(128) void sample_proj_kernel(
    const long long* __restrict__ num_rows, const long long* __restrict__ num_cols,
    const float* __restrict__ feat, const float* __restrict__ boxes,
    const float* __restrict__ proj_w, const float* __restrict__ proj_b,
    const float* __restrict__ bbox_w, const float* __restrict__ bbox_b,
    const float* __restrict__ bboxln_w, const float* __restrict__ bboxln_b,
    float* __restrict__ fp_ws) {
  const int bi = blockIdx.y;
  const int R = (int)num_rows[bi], C = (int)num_cols[bi];
  const int GC = 2 * C;
  const int NP = (2 * R) * GC;
  const int tile0 = blockIdx.x * 16;
  if (tile0 >= NP) return;

  __shared__ float Asub[16][260];     // sampled features, 16 points x 256 ch
  __shared__ float PosLn[16][260];    // positional embedding (layernormed)
  __shared__ int   addr4[16][4];
  __shared__ float w4[16][4];
  __shared__ float gxy[16][2];
  __shared__ int   pidx[16];
  __shared__ float red[16][8][2];

  const int t = threadIdx.x;

  // ---- per-point geometry (threads 0..15) ----
  if (t < 16) {
    int p = tile0 + t;
    if (p >= NP) p = NP - 1;
    pidx[t] = p;
    int py = p / GC, px = p - py * GC;
    int r = py >> 1, i = py & 1;
    int c = px >> 1, j = px & 1;
    int corner = i * 2 + j;
    size_t bidx = ((((size_t)bi * MAXR + r) * MAXC + c) * 4 + corner) * 2;
    float gx = boxes[bidx + 0] * 0.25f;     // SCALE * SCALE
    float gy = boxes[bidx + 1] * 0.25f;
    float gxn = (gx * (1.0f / WW) - 0.5f) * 2.0f;
    float gyn = (gy * (1.0f / HH) - 0.5f) * 2.0f;
    gxy[t][0] = gxn; gxy[t][1] = gyn;
    float x = (gxn + 1.0f) * (WW * 0.5f) - 0.5f;
    float y = (gyn + 1.0f) * (HH * 0.5f) - 0.5f;
    float x0 = floorf(x), y0 = floorf(y);
    float wx1 = x - x0, wx0 = 1.0f - wx1;
    float wy1 = y - y0, wy0 = 1.0f - wy1;
    int xi0 = (int)x0, yi0 = (int)y0;
#pragma unroll
    for (int tap = 0; tap < 4; ++tap) {
      int xi = xi0 + (tap & 1), yi = yi0 + (tap >> 1);
      bool valid = (xi >= 0) && (xi < WW) && (yi >= 0) && (yi < HH);
      int xc = min(max(xi, 0), WW - 1), yc = min(max(yi, 0), HH - 1);
      addr4[t][tap] = yc * WW + xc;
      float wt = ((tap & 1) ? wx1 : wx0) * ((tap >> 1) ? wy1 : wy0);
      w4[t][tap] = valid ? wt : 0.0f;
    }
  }
  __syncthreads();

  const int m = t >> 3;          // 0..15 point within tile
  const int sub = t & 7;         // 0..7 channel-chunk

  // ---- bilinear sample 16 points x 256 channels into LDS ----
  {
    const float* fch = feat + (size_t)bi * CIN * HWSZ;
    int a0 = addr4[m][0], a1 = addr4[m][1], a2 = addr4[m][2], a3 = addr4[m][3];
    float w0 = w4[m][0], w1 = w4[m][1], w2 = w4[m][2], w3 = w4[m][3];
#pragma unroll 1
    for (int kk = 0; kk < 32; ++kk) {
      int k = sub * 32 + kk;
      const float* f = fch + (size_t)k * HWSZ;
      Asub[m][k] = w0 * f[a0] + w1 * f[a1] + w2 * f[a2] + w3 * f[a3];
    }
  }

  // ---- positional embedding (K=2 GEMM) + layernorm over D ----
  {
    float gxn = gxy[m][0], gyn = gxy[m][1];
    float s = 0.f, s2 = 0.f;
#pragma unroll 1
    for (int dd = 0; dd < 32; ++dd) {
      int d = sub * 32 + dd;
      float v = gxn * bbox_w[d * 2 + 0] + gyn * bbox_w[d * 2 + 1] + bbox_b[d];
      PosLn[m][d] = v;
      s += v; s2 += v * v;
    }
    red[m][sub][0] = s; red[m][sub][1] = s2;
    __syncthreads();
    float ts = 0.f, ts2 = 0.f;
#pragma unroll
    for (int q = 0; q < 8; ++q) { ts += red[m][q][0]; ts2 += red[m][q][1]; }
    float mean = ts * (1.0f / 256.0f);
    float var = ts2 * (1.0f / 256.0f) - mean * mean;
    float rstd = rsqrtf(var + 1e-5f);
#pragma unroll 1
    for (int dd = 0; dd < 32; ++dd) {
      int d = sub * 32 + dd;
      PosLn[m][d] = (PosLn[m][d] - mean) * rstd * bboxln_w[d] + bboxln_b[d];
    }
  }
  __syncthreads();

  // ---- projection GEMM: 16 points x 256 outputs, K=256, fp32 WMMA ----
  const int lane = t & 31;
  const int wv = t >> 5;                 // wave 0..3 -> 64 output channels each
  const int mrow = lane & 15;
  const int khalf = (lane >> 4) << 1;    // lanes 16-31 hold K+2,K+3
  v8f acc[4];
#pragma unroll
  for (int nt = 0; nt < 4; ++nt) acc[nt] = (v8f){0, 0, 0, 0, 0, 0, 0, 0};

  for (int k = 0; k < CIN; k += 4) {
    int kb = k + khalf;
    v2f a; a.x = Asub[mrow][kb]; a.y = Asub[mrow][kb + 1];
#pragma unroll
    for (int nt = 0; nt < 4; ++nt) {
      int n = wv * 64 + nt * 16 + mrow;
      const float* bp = proj_w + (size_t)n * CIN + kb;  // B[k][n] = proj_w[n][k]
      v2f b; b.x = bp[0]; b.y = bp[1];
      acc[nt] = __builtin_amdgcn_wmma_f32_16x16x4_f32(
          false, a, false, b, (short)0, acc[nt], false, false);
    }
  }

  // ---- epilogue: fp = proj + proj_b + pos_ln -> fp_ws ----
  const int mbase = (lane >> 4) * 8;   // C layout: vgpr v -> M=v (+8 for hi lanes)
#pragma unroll
  for (int v = 0; v < 8; ++v) {
    int mm = mbase + v;
    int p = pidx[mm];
#pragma unroll
    for (int nt = 0; nt < 4; ++nt) {
      int n = wv * 64 + nt * 16 + (lane & 15);
      float val = acc[nt][v] + proj_b[n] + PosLn[mm][n];
      fp_ws[((size_t)bi * GMAX + p) * DD + n] = val;
    }
  }
}

// ---------------------------------------------------------------------------
// Kernel 2: 2x2/stride-2 conv as GEMM (K=1024, fp32 WMMA) + layernorm + scatter
// ---------------------------------------------------------------------------
__global__ __launch_bounds__(128) void conv_ln_kernel(
    const long long* __restrict__ num_rows, const long long* __restrict__ num_cols,
    const float* __restrict__ conv_w, const float* __restrict__ conv_b,
    const float* __restrict__ addln_w, const float* __restrict__ addln_b,
    const float* __restrict__ fp_ws, float* __restrict__ out) {
  const int bi = blockIdx.y;
  const int R = (int)num_rows[bi], C = (int)num_cols[bi];
  const int GC = 2 * C;
  const int NPos = R * C;
  const int tile0 = blockIdx.x * 16;
  if (tile0 >= NPos) return;

  __shared__ float A2[16][1028];      // k = ij*256 + e
  __shared__ float Cres[16][260];
  __shared__ int   prc[16][2];
  __shared__ float red[16][8][2];

  const int t = threadIdx.x;
  if (t < 16) {
    int p = tile0 + t;
    if (p >= NPos) p = NPos - 1;
    int r = p / C, c = p - r * C;
    prc[t][0] = r; prc[t][1] = c;
  }
  __syncthreads();

  const int m = t >> 3, sub = t & 7;

  // ---- stage A tile: 16 positions x (4 taps x 256 ch) from fp_ws ----
  {
    const int ij = sub >> 1;     // tap 0..3 (i = ij>>1, j = ij&1)
    const int half = sub & 1;    // channel half
    int r = prc[m][0], c = prc[m][1];
    int i = ij >> 1, j = ij & 1;
    const float* src = fp_ws +
        ((size_t)bi * GMAX + (size_t)(2 * r + i) * GC + (2 * c + j)) * DD + half * 128;
    float* dst = &A2[m][ij * 256 + half * 128];
#pragma unroll 1
    for (int e = 0; e < 128; e += 4) {
      float4 v = *(const float4*)(src + e);
      *(float4*)(dst + e) = v;
    }
  }
  __syncthreads();

  // ---- conv GEMM: K = 1024, fp32 WMMA; B[k][n] = conv_w[n*1024 + e*4 + ij] ----
  const int lane = t & 31;
  const int wv = t >> 5;
  const int mrow = lane & 15;
  const int khalf = (lane >> 4) << 1;
  v8f acc[4];
#pragma unroll
  for (int nt = 0; nt < 4; ++nt) acc[nt] = (v8f){0, 0, 0, 0, 0, 0, 0, 0};

  for (int k = 0; k < 1024; k += 4) {
    int kb = k + khalf;
    v2f a; a.x = A2[mrow][kb]; a.y = A2[mrow][kb + 1];
    int e = kb & 255, ij = kb >> 8;
#pragma unroll
    for (int nt = 0; nt < 4; ++nt) {
      int n = wv * 64 + nt * 16 + mrow;
      const float* bp = conv_w + (size_t)n * 1024 + (size_t)e * 4 + ij;
      v2f b; b.x = bp[0]; b.y = bp[4];   // k+1 -> e+1 -> +4 floats
      acc[nt] = __builtin_amdgcn_wmma_f32_16x16x4_f32(
          false, a, false, b, (short)0, acc[nt], false, false);
    }
  }

  // ---- spill to LDS (+ conv_b) for cross-wave layernorm ----
  const int mbase = (lane >> 4) * 8;
#pragma unroll
  for (int v = 0; v < 8; ++v) {
    int mm = mbase + v;
#pragma unroll
    for (int nt = 0; nt < 4; ++nt) {
      int n = wv * 64 + nt * 16 + (lane & 15);
      Cres[mm][n] = acc[nt][v] + conv_b[n];
    }
  }
  __syncthreads();

  // ---- layernorm over D and scatter to (B, D, 64, 64) ----
  {
    float s = 0.f, s2 = 0.f;
#pragma unroll 1
    for (int dd = 0; dd < 32; ++dd) {
      float x = Cres[m][sub * 32 + dd];
      s += x; s2 += x * x;
    }
    red[m][sub][0] = s; red[m][sub][1] = s2;
    __syncthreads();
    float ts = 0.f, ts2 = 0.f;
#pragma unroll
    for (int q = 0; q < 8; ++q) { ts += red[m][q][0]; ts2 += red[m][q][1]; }
    float mean = ts * (1.0f / 256.0f);
    float var = ts2 * (1.0f / 256.0f) - mean * mean;
    float rstd = rsqrtf(var + 1e-5f);
    if (tile0 + m < NPos) {
      int r = prc[m][0], c = prc[m][1];
#pragma unroll 1
      for (int dd = 0; dd < 32; ++dd) {
        int d = sub * 32 + dd;
        float y = (Cres[m][d] - mean) * rstd * addln_w[d] + addln_b[d];
        out[(((size_t)bi * DD + d) * MAXR + r) * MAXC + c] = y;
      }
    }
  }
}

// ---------------------------------------------------------------------------
extern "C" void kernel_launch(void* const* d_in, const int* in_sizes, int n_in,
                              void* d_out, int out_size, void* d_ws, size_t ws_size,
                              hipStream_t stream) {
  const long long* num_rows = (const long long*)d_in[0];
  const long long* num_cols = (const long long*)d_in[1];
  const float* feat    = (const float*)d_in[2];
  const float* boxes   = (const float*)d_in[3];
  const float* proj_w  = (const float*)d_in[4];
  const float* proj_b  = (const float*)d_in[5];
  const float* bbox_w  = (const float*)d_in[6];
  const float* bbox_b  = (const float*)d_in[7];
  const float* bboxln_w = (const float*)d_in[8];
  const float* bboxln_b = (const float*)d_in[9];
  const float* conv_w  = (const float*)d_in[10];
  const float* conv_b  = (const float*)d_in[11];
  const float* addln_w = (const float*)d_in[12];
  const float* addln_b = (const float*)d_in[13];

  float* out  = (float*)d_out;                              // (8,256,64,64)
  float* mask = out + (size_t)NB * DD * MAXR * MAXC;        // (8,64,64)
  float* fp_ws = (float*)d_ws;                              // [8][16384][256] f32

  {
    size_t total = (size_t)NB * DD * MAXR * MAXC;
    int blocks = (int)((total + 255) / 256);
    init_out_kernel<<<blocks, 256, 0, stream>>>(num_rows, num_cols, out, mask);
  }
  {
    dim3 grid(GMAX / 16, NB);      // 1024 x 8 tiles of 16 points
    sample_proj_kernel<<<grid, 128, 0, stream>>>(
        num_rows, num_cols, feat, boxes, proj_w, proj_b,
        bbox_w, bbox_b, bboxln_w, bboxln_b, fp_ws);
  }
  {
    dim3 grid((MAXR * MAXC) / 16, NB);   // 256 x 8 tiles of 16 positions
    conv_ln_kernel<<<grid, 128, 0, stream>>>(
        num_rows, num_cols, conv_w, conv_b, addln_w, addln_b, fp_ws, out);
  }
}